// Spectroformer_70866960384422
// MI455X (gfx1250) — compile-verified
//
#include <hip/hip_runtime.h>
#include <math.h>

// ---------------------------------------------------------------------------
// MQCT block for MI455X (gfx1250, wave32).
// All GEMM-shaped work (1x1 convs, attention score split-K, attention apply)
// uses V_WMMA_F32_16X16X4_F32 (fp32 in/out; every GEMM here is HBM-bound at
// 23.3 TB/s so fp32 WMMA already saturates bandwidth). Hot loops are fully
// branchless (address clamping for OOB rows/cols, single value-masked K-tail
// iteration) and use base-pointer + 32-bit offsets so the backend can emit
// SGPR-base+VGPR-offset global loads with one 32-bit add per pointer/iter.
// FFTs are 256-pt LDS Stockham kernels. Workspace: bump arena w/ phase reuse,
// peak ~610 MB (assumed available in d_ws).
// ---------------------------------------------------------------------------

typedef float v2f __attribute__((ext_vector_type(2)));
typedef float v8f __attribute__((ext_vector_type(8)));

static constexpr int  Bn    = 2;
static constexpr int  C     = 96;
static constexpr int  Hh    = 256;
static constexpr int  Wd    = 256;
static constexpr int  HEADS = 8;
static constexpr int  HD    = 12;
static constexpr int  HID   = 255;
static constexpr long HW    = (long)Hh * Wd;   // 65536
static constexpr int  WF    = Wd / 2 + 1;      // 129
static constexpr long HF    = (long)Hh * WF;   // 33024
static constexpr float PI_F = 3.14159265358979323846f;

__device__ __forceinline__ float geluf(float x) {
  return 0.5f * x * (1.0f + erff(x * 0.70710678118654752f));
}

__device__ __forceinline__ v8f wmma4(v2f a, v2f b, v8f c) {
  return __builtin_amdgcn_wmma_f32_16x16x4_f32(false, a, false, b,
                                               (short)0, c, false, false);
}

// ------------------------- generic 1x1-conv GEMM ---------------------------
// Out[b, m, n] = gelu?( sum_k W[m,k] * In[b, k, n] ) + Resid[b, m, n]
// One wave: M=16 x N=64 (4 accum tiles). Block = 8 waves covering N=512.
__global__ __launch_bounds__(256)
void gemm1x1(const float* __restrict__ Wt, const float* __restrict__ In,
             float* Out, const float* Resid,
             int M, int K, long N,
             long inBS, long outBS, long resBS, int doGelu)
{
  const int lane = threadIdx.x & 31;
  const int wv   = threadIdx.x >> 5;
  const int hi   = lane >> 4;       // half-wave: K offset of the A/B frags
  const int lm   = lane & 15;
  const int b    = blockIdx.z;
  const int mB   = blockIdx.y * 16;
  const long nB  = (long)blockIdx.x * 512 + (long)wv * 64;

  const float* in  = In  + (long)b * inBS;
  float*       out = Out + (long)b * outBS;

  const int  m  = mB + lm;
  const long mc = (m < M) ? m : (M - 1);       // clamped A row

  unsigned nc[4];
#pragma unroll
  for (int j = 0; j < 4; ++j) {
    const long n = nB + j * 16 + lm;
    nc[j] = (unsigned)((n < N) ? n : (N - 1)); // clamped B column
  }

  const unsigned uN    = (unsigned)N;
  const unsigned stepB = 4u * uN;
  unsigned wOff = (unsigned)(mc * K + 2 * hi);
  unsigned o0 = 2u * (unsigned)hi * uN + nc[0];
  unsigned o1 = 2u * (unsigned)hi * uN + nc[1];
  unsigned o2 = 2u * (unsigned)hi * uN + nc[2];
  unsigned o3 = 2u * (unsigned)hi * uN + nc[3];

  v8f acc0 = v8f{}, acc1 = v8f{}, acc2 = v8f{}, acc3 = v8f{};

  const int K4 = K & ~3;
  for (int k0 = 0; k0 < K4; k0 += 4) {
    __builtin_prefetch(in + (o0 + 4 * stepB), 0, 0);   // global_prefetch_b8
    __builtin_prefetch(in + (o2 + 4 * stepB), 0, 0);
    v2f a;  a.x  = Wt[wOff]; a.y  = Wt[wOff + 1];
    v2f b0; b0.x = in[o0];   b0.y = in[o0 + uN];
    acc0 = wmma4(a, b0, acc0);
    v2f b1; b1.x = in[o1];   b1.y = in[o1 + uN];
    acc1 = wmma4(a, b1, acc1);
    v2f b2; b2.x = in[o2];   b2.y = in[o2 + uN];
    acc2 = wmma4(a, b2, acc2);
    v2f b3; b3.x = in[o3];   b3.y = in[o3 + uN];
    acc3 = wmma4(a, b3, acc3);
    wOff += 4; o0 += stepB; o1 += stepB; o2 += stepB; o3 += stepB;
  }

  if (K4 < K) {                                  // K tail (only K=255 here)
    const int  ka  = K4 + 2 * hi;
    const float mx = (ka     < K) ? 1.0f : 0.0f;
    const float my = (ka + 1 < K) ? 1.0f : 0.0f;
    const long kax = (ka     < K) ? ka     : (K - 1);
    const long kay = (ka + 1 < K) ? ka + 1 : (K - 1);
    v2f a;
    a.x = Wt[mc * K + kax] * mx;
    a.y = Wt[mc * K + kay] * my;
    v8f* accs[4] = {&acc0, &acc1, &acc2, &acc3};
#pragma unroll
    for (int j = 0; j < 4; ++j) {
      v2f bb;
      bb.x = in[kax * N + nc[j]] * mx;
      bb.y = in[kay * N + nc[j]] * my;
      *accs[j] = wmma4(a, bb, *accs[j]);
    }
  }

  const v8f accs[4] = {acc0, acc1, acc2, acc3};
#pragma unroll
  for (int j = 0; j < 4; ++j) {
    const long n = nB + j * 16 + lm;
    if (n >= N) continue;
#pragma unroll
    for (int r = 0; r < 8; ++r) {
      const int mo = mB + r + 8 * hi;
      if (mo < M) {
        float v = accs[j][r];
        if (doGelu) v = geluf(v);
        if (Resid)  v += Resid[(long)b * resBS + (long)mo * N + n];
        out[(long)mo * N + n] = v;
      }
    }
  }
}

// --------------------------- depthwise 3x3 SAME ----------------------------
__global__ __launch_bounds__(256)
void dwconv3x3(const float* __restrict__ in, const float* __restrict__ w,
               float* __restrict__ out, int Cc, long total)
{
  const long i = (long)blockIdx.x * 256 + threadIdx.x;
  if (i >= total) return;
  const int  x  = (int)(i & (Wd - 1));
  const int  y  = (int)((i >> 8) & (Hh - 1));
  const long pl = i >> 16;                 // b*Cc + c
  const long cc = pl % Cc;
  const float* p  = in + (pl << 16);
  const float* wk = w + cc * 9;
  float s = 0.0f;
#pragma unroll
  for (int ky = -1; ky <= 1; ++ky) {
    const int yy = y + ky;
    if (yy < 0 || yy >= Hh) continue;
#pragma unroll
    for (int kx = -1; kx <= 1; ++kx) {
      const int xx = x + kx;
      if (xx < 0 || xx >= Wd) continue;
      s += wk[(ky + 1) * 3 + (kx + 1)] * p[(long)yy * Wd + xx];
    }
  }
  out[i] = s;
}

// ------------------------- LayerNorm over channels -------------------------
__global__ __launch_bounds__(256)
void layernorm_c(const float* __restrict__ x, const float* __restrict__ g,
                 const float* __restrict__ bta, float* __restrict__ y)
{
  const long i = (long)blockIdx.x * 256 + threadIdx.x;  // over Bn*HW
  if (i >= (long)Bn * HW) return;
  const long b = i >> 16;
  const long p = i & (HW - 1);
  const float* xp = x + b * C * HW + p;
  float s = 0.0f, s2 = 0.0f;
  for (int c = 0; c < C; ++c) { const float v = xp[(long)c * HW]; s += v; s2 += v * v; }
  const float mu  = s * (1.0f / C);
  const float var = s2 * (1.0f / C) - mu * mu;
  const float inv = rsqrtf(var + 1e-5f);
  float* yp = y + b * C * HW + p;
  for (int c = 0; c < C; ++c) {
    const float v = xp[(long)c * HW];
    yp[(long)c * HW] = (v - mu) * inv * g[c] + bta[c];
  }
}

// ------------------- l2 row norms (over n=HW, per channel) -----------------
__global__ __launch_bounds__(256)
void rownorm(const float* __restrict__ base, long bstride, float* __restrict__ norms)
{
  const int r = blockIdx.x;          // Bn*C rows
  const int b = r / C, c = r % C;
  const float* p = base + (long)b * bstride + (long)c * HW;
  float s = 0.0f;
  for (long i = threadIdx.x; i < HW; i += 256) { const float v = p[i]; s += v * v; }
  __shared__ float red[256];
  red[threadIdx.x] = s; __syncthreads();
  for (int st = 128; st > 0; st >>= 1) {
    if (threadIdx.x < st) red[threadIdx.x] += red[threadIdx.x + st];
    __syncthreads();
  }
  if (threadIdx.x == 0) norms[r] = fmaxf(sqrtf(red[0]), 1e-12f);
}

// ------------- attention scores: split-K WMMA + softmax in-block -----------
// S[b,h,d,e] = softmax_e( (q_n[d,:] . k_n[e,:]) * temp[h] ); K = HW split
// over 8 waves, each accumulating 16x16 partials with WMMA; LDS reduction.
// Rows d,e >= HD use a clamped (valid) pointer; their tile entries are
// simply never read back, so the loop needs no guards at all.
__global__ __launch_bounds__(256)
void attn_scores(const float* __restrict__ qbase, long qbs,
                 const float* __restrict__ kbase, long kbs,
                 const float* __restrict__ normq, const float* __restrict__ normk,
                 const float* __restrict__ temp, float* __restrict__ Sg)
{
  const int bh = blockIdx.x;               // Bn*HEADS = 16
  const int b = bh / HEADS, h = bh % HEADS;
  const int lane = threadIdx.x & 31, wv = threadIdx.x >> 5;
  const int hi = lane >> 4, lm = lane & 15;
  const int rowc = (lm < HD) ? lm : (HD - 1);
  const float* q = qbase + (long)b * qbs + (long)h * HD * HW;
  const float* k = kbase + (long)b * kbs + (long)h * HD * HW;

  unsigned off = (unsigned)((long)rowc * HW + (long)wv * (HW / 8) + 2 * hi);
  v8f acc = v8f{};
  for (long t = 0; t < HW / 8; t += 4) {
    __builtin_prefetch(q + off + 64, 0, 0);
    const v2f a  = *(const v2f*)(q + off);   // 8B aligned
    const v2f bb = *(const v2f*)(k + off);
    acc = wmma4(a, bb, acc);
    off += 4;
  }

  __shared__ float sacc[8 * 256];
#pragma unroll
  for (int r = 0; r < 8; ++r)
    sacc[wv * 256 + (r + 8 * hi) * 16 + lm] = acc[r];
  __syncthreads();

  __shared__ float sS[HD * HD];
  const int t = threadIdx.x;
  if (t < HD * HD) {
    const int d = t / HD, e = t % HD;
    float s = 0.0f;
    for (int w2 = 0; w2 < 8; ++w2) s += sacc[w2 * 256 + d * 16 + e];
    s = s / (normq[b * C + h * HD + d] * normk[b * C + h * HD + e]) * temp[h];
    sS[t] = s;
  }
  __syncthreads();
  if (t < HD) {
    float mx = -1e30f;
    for (int e = 0; e < HD; ++e) mx = fmaxf(mx, sS[t * HD + e]);
    float ex[HD]; float sum = 0.0f;
    for (int e = 0; e < HD; ++e) { ex[e] = __expf(sS[t * HD + e] - mx); sum += ex[e]; }
    const float inv = 1.0f / sum;
    for (int e = 0; e < HD; ++e) Sg[bh * 144 + t * HD + e] = ex[e] * inv;
  }
}

// ---------------- attention apply: out[d,n] = sum_e A[d,e] v[e,n] ----------
// A fragment is zero-padded for d,e >= HD (so clamped-address garbage B rows
// are annihilated); inner loop is unguarded loads + WMMA.
__global__ __launch_bounds__(256)
void attn_apply(const float* __restrict__ Sg, const float* __restrict__ vbase,
                long vbs, float* __restrict__ outbase)
{
  const int bh = blockIdx.y;
  const int b = bh / HEADS, h = bh % HEADS;
  const int lane = threadIdx.x & 31, wv = threadIdx.x >> 5;
  const int hi = lane >> 4, lm = lane & 15;
  const float* v = vbase + (long)b * vbs + (long)h * HD * HW;
  const float* A = Sg + bh * 144;

  v2f a[4];
  unsigned vA[4], vB[4];
#pragma unroll
  for (int kk = 0; kk < 4; ++kk) {
    const int e0 = kk * 4 + 2 * hi;
    a[kk].x = (lm < HD && e0     < HD) ? A[lm * HD + e0]     : 0.0f;
    a[kk].y = (lm < HD && e0 + 1 < HD) ? A[lm * HD + e0 + 1] : 0.0f;
    const unsigned e0c = (unsigned)((e0     < HD) ? e0     : (HD - 1));
    const unsigned e1c = (unsigned)((e0 + 1 < HD) ? e0 + 1 : (HD - 1));
    vA[kk] = e0c * (unsigned)HW;
    vB[kk] = e1c * (unsigned)HW;
  }

  const long nB = (long)blockIdx.x * 512 + (long)wv * 64;
  float* outp = outbase + (long)b * C * HW + (long)h * HD * HW;
#pragma unroll
  for (int j = 0; j < 4; ++j) {
    const unsigned n = (unsigned)(nB + j * 16 + lm);
    v8f acc = v8f{};
#pragma unroll
    for (int kk = 0; kk < 4; ++kk) {
      v2f bb;
      bb.x = v[vA[kk] + n];
      bb.y = v[vB[kk] + n];
      acc = wmma4(a[kk], bb, acc);
    }
#pragma unroll
    for (int r = 0; r < 8; ++r) {
      const int d = r + 8 * hi;
      if (d < HD) outp[(unsigned)d * (unsigned)HW + n] = acc[r];
    }
  }
}

// ------------------------- 256-pt Stockham FFT in LDS ----------------------
struct cpx { float x, y; };
__device__ __forceinline__ cpx cmul(cpx a, cpx b) {
  return cpx{a.x * b.x - a.y * b.y, a.x * b.y + a.y * b.x};
}

// sign = -1 forward, +1 inverse. Result ends in buf0 (8 stages).
__device__ void fft256(cpx* buf0, cpx* buf1, int tid, float sign)
{
  cpx* src = buf0; cpx* dst = buf1;
#pragma unroll
  for (int Ns = 1; Ns < 256; Ns <<= 1) {
    const int j  = tid;
    const int jl = j & (Ns - 1);
    cpx u = src[j];
    cpx v = src[j + 128];
    const float ang = sign * PI_F * (float)jl / (float)Ns;
    float sw, cw; __sincosf(ang, &sw, &cw);
    v = cmul(v, cpx{cw, sw});
    const int d = ((j - jl) << 1) + jl;
    dst[d]      = cpx{u.x + v.x, u.y + v.y};
    dst[d + Ns] = cpx{u.x - v.x, u.y - v.y};
    __syncthreads();
    cpx* tsw = src; src = dst; dst = tsw;
  }
}

// rows: real 256 -> bins 0..128 into f (b, [re:0..95 | im:96..191], h, wf)
__global__ __launch_bounds__(128)
void fft_row_fwd(const float* __restrict__ y, float* __restrict__ f)
{
  __shared__ cpx b0[256], b1[256];
  const int row = blockIdx.x;              // Bn*C*Hh
  const int h   = row & 255;
  const int bc  = row >> 8;                // b*C + c
  const int b   = bc / C, c = bc % C;
  const float* in = y + (long)bc * HW + (long)h * Wd;
  const int t = threadIdx.x;
  b0[t]       = cpx{in[t],       0.0f};
  b0[t + 128] = cpx{in[t + 128], 0.0f};
  __syncthreads();
  fft256(b0, b1, t, -1.0f);
  float* fr = f + ((long)(b * 192 + c)      * Hh + h) * WF;
  float* fi = f + ((long)(b * 192 + 96 + c) * Hh + h) * WF;
  fr[t] = b0[t].x * 0.0625f;  fi[t] = b0[t].y * 0.0625f;
  if (t == 0) { fr[128] = b0[128].x * 0.0625f; fi[128] = b0[128].y * 0.0625f; }
}

// columns: complex 256-pt along h, in place on f planes (re plane c, im 96+c)
__global__ __launch_bounds__(128)
void fft_col(float* __restrict__ f, float sign)
{
  __shared__ cpx b0[256], b1[256];
  const int col = blockIdx.x;              // Bn*C*WF
  const int wf  = col % WF;
  const int bc  = col / WF;
  const int b   = bc / C, c = bc % C;
  float* fr = f + ((long)(b * 192 + c)      * Hh) * WF + wf;
  float* fi = f + ((long)(b * 192 + 96 + c) * Hh) * WF + wf;
  const int t = threadIdx.x;
  b0[t]       = cpx{fr[(long)t * WF],         fi[(long)t * WF]};
  b0[t + 128] = cpx{fr[(long)(t + 128) * WF], fi[(long)(t + 128) * WF]};
  __syncthreads();
  fft256(b0, b1, t, sign);
  fr[(long)t * WF]         = b0[t].x * 0.0625f;
  fi[(long)t * WF]         = b0[t].y * 0.0625f;
  fr[(long)(t + 128) * WF] = b0[t + 128].x * 0.0625f;
  fi[(long)(t + 128) * WF] = b0[t + 128].y * 0.0625f;
}

// rows inverse: 129 bins + Hermitian symmetry -> 256 real samples (q2)
__global__ __launch_bounds__(128)
void fft_row_inv_c2r(const float* __restrict__ f, float* __restrict__ q2)
{
  __shared__ cpx b0[256], b1[256];
  const int row = blockIdx.x;
  const int h   = row & 255;
  const int bc  = row >> 8;
  const int b   = bc / C, c = bc % C;
  const float* fr = f + ((long)(b * 192 + c)      * Hh + h) * WF;
  const float* fi = f + ((long)(b * 192 + 96 + c) * Hh + h) * WF;
  const int t = threadIdx.x;
  b0[t] = cpx{fr[t], fi[t]};
  const int p = t + 128;
  if (p <= 128) b0[p] = cpx{fr[p], fi[p]};
  else          b0[p] = cpx{fr[256 - p], -fi[256 - p]};
  __syncthreads();
  fft256(b0, b1, t, +1.0f);
  float* out = q2 + (long)bc * HW + (long)h * Wd;
  out[t]       = b0[t].x       * 0.0625f;
  out[t + 128] = b0[t + 128].x * 0.0625f;
}

// --------------------------- GDFN gating -----------------------------------
__global__ __launch_bounds__(256)
void gdfn_gate(const float* __restrict__ g2, float* __restrict__ gg)
{
  const long i = (long)blockIdx.x * 256 + threadIdx.x;  // Bn*HID*HW
  if (i >= (long)Bn * HID * HW) return;
  const long p  = i & (HW - 1);
  const long cb = i >> 16;             // b*HID + c
  const long b  = cb / HID, c = cb % HID;
  const float* base = g2 + b * (long)(2 * HID) * HW;
  const float y1 = base[c * HW + p];
  const float y2 = base[(c + HID) * HW + p];
  gg[i] = geluf(y1) * y2;
}

// ===========================================================================
extern "C" void kernel_launch(void* const* d_in, const int* in_sizes, int n_in,
                              void* d_out, int out_size, void* d_ws, size_t ws_size,
                              hipStream_t stream)
{
  (void)in_sizes; (void)n_in; (void)out_size; (void)ws_size;
  const float* x       = (const float*)d_in[0];
  const float* ln1_g   = (const float*)d_in[1];
  const float* ln1_b   = (const float*)d_in[2];
  const float* qkv1_w  = (const float*)d_in[3];
  const float* qkv1_dw = (const float*)d_in[4];
  const float* proj1_w = (const float*)d_in[5];
  const float* fdfp_w1 = (const float*)d_in[6];
  const float* fdfp_w2 = (const float*)d_in[7];
  const float* kv2_w   = (const float*)d_in[8];
  const float* kv2_dw  = (const float*)d_in[9];
  const float* proj2_w = (const float*)d_in[10];
  const float* temp    = (const float*)d_in[11];
  const float* ln2_g   = (const float*)d_in[12];
  const float* ln2_b   = (const float*)d_in[13];
  const float* gin_w   = (const float*)d_in[14];
  const float* gdw_w   = (const float*)d_in[15];
  const float* gout_w  = (const float*)d_in[16];
  float* out = (float*)d_out;
  float* ws  = (float*)d_ws;

  const long SZ_X = (long)Bn * C * HW;      // 12.58M floats
  const long SZ_F = (long)Bn * 192 * HF;    // 12.68M floats

  // ---- MQCA arena ----
  float* y    = ws;                         // LN1 output (kept: FFT input)
  float* t1   = ws + SZ_X;                  // qkv conv out (3C planes)
  float* t2   = t1 + 3 * SZ_X;              // qkv dwconv out (q1|k1|v1)
  float* oatt = t1;                         // attn out (reuses dead t1)
  float* o1   = t1 + SZ_X;                  // proj1 out
  float* f    = t2;                         // FFT planes (reuses dead t2)
  float* fm   = f  + SZ_F;
  float* fo   = fm + SZ_F;
  float* q2   = fo + SZ_F;
  float* kv   = q2 + SZ_X;                  // 2C planes
  float* kvd  = kv + 2 * SZ_X;              // 2C planes
  float* smal = kvd + 2 * SZ_X;
  float* normq = smal, * normk = smal + 256, * Sg = smal + 512;  // 16*144

  // ---- GDFN arena (overlays MQCA region; valid once d_out holds x1) ----
  float* y2 = ws;                           // LN2 out
  float* g1 = ws + SZ_X;                    // 2*HID planes
  float* g2 = g1 + (long)Bn * 2 * HID * HW; // 2*HID planes
  float* gg = g1;                           // gated (reuses dead g1)

  const long bsX  = (long)C * HW;           // (B,C,HW) batch stride
  const long bs3X = 3 * bsX;
  const long bs2X = 2 * bsX;
  const long bsF  = 192 * HF;

  // ---------------- Stage 1: spatial channel attention ----------------
  layernorm_c<<<(unsigned)((Bn * HW) / 256), 256, 0, stream>>>(x, ln1_g, ln1_b, y);

  gemm1x1<<<dim3(HW / 512, 288 / 16, Bn), 256, 0, stream>>>(
      qkv1_w, y, t1, nullptr, 288, 96, HW, bsX, bs3X, 0, 0);
  dwconv3x3<<<(unsigned)((Bn * 288 * HW) / 256), 256, 0, stream>>>(
      t1, qkv1_dw, t2, 288, (long)Bn * 288 * HW);

  rownorm<<<Bn * C, 256, 0, stream>>>(t2,           bs3X, normq);  // q1
  rownorm<<<Bn * C, 256, 0, stream>>>(t2 + 96 * HW, bs3X, normk);  // k1
  attn_scores<<<Bn * HEADS, 256, 0, stream>>>(
      t2, bs3X, t2 + 96 * HW, bs3X, normq, normk, temp, Sg);
  attn_apply<<<dim3(HW / 512, Bn * HEADS), 256, 0, stream>>>(
      Sg, t2 + 192 * HW, bs3X, oatt);

  gemm1x1<<<dim3(HW / 512, 96 / 16, Bn), 256, 0, stream>>>(
      proj1_w, oatt, o1, nullptr, 96, 96, HW, bsX, bsX, 0, 0);

  // ---------------- Stage 2: FDFP frequency branch ----------------
  fft_row_fwd<<<(unsigned)(Bn * C * Hh), 128, 0, stream>>>(y, f);
  fft_col<<<(unsigned)(Bn * C * WF), 128, 0, stream>>>(f, -1.0f);

  gemm1x1<<<dim3((HF + 511) / 512, 192 / 16, Bn), 256, 0, stream>>>(
      fdfp_w1, f, fm, nullptr, 192, 192, HF, bsF, bsF, 0, 1 /*gelu*/);
  gemm1x1<<<dim3((HF + 511) / 512, 192 / 16, Bn), 256, 0, stream>>>(
      fdfp_w2, fm, fo, nullptr, 192, 192, HF, bsF, bsF, 0, 0);

  fft_col<<<(unsigned)(Bn * C * WF), 128, 0, stream>>>(fo, +1.0f);
  fft_row_inv_c2r<<<(unsigned)(Bn * C * Hh), 128, 0, stream>>>(fo, q2);

  gemm1x1<<<dim3(HW / 512, 192 / 16, Bn), 256, 0, stream>>>(
      kv2_w, o1, kv, nullptr, 192, 96, HW, bsX, bs2X, 0, 0);
  dwconv3x3<<<(unsigned)((Bn * 192 * HW) / 256), 256, 0, stream>>>(
      kv, kv2_dw, kvd, 192, (long)Bn * 192 * HW);

  rownorm<<<Bn * C, 256, 0, stream>>>(q2,  bsX,  normq);   // q2
  rownorm<<<Bn * C, 256, 0, stream>>>(kvd, bs2X, normk);   // k2
  attn_scores<<<Bn * HEADS, 256, 0, stream>>>(
      q2, bsX, kvd, bs2X, normq, normk, temp, Sg);
  attn_apply<<<dim3(HW / 512, Bn * HEADS), 256, 0, stream>>>(
      Sg, kvd + 96 * HW, bs2X, oatt);

  // proj2 + residual: d_out = x + conv1x1(attn2, proj2_w)
  gemm1x1<<<dim3(HW / 512, 96 / 16, Bn), 256, 0, stream>>>(
      proj2_w, oatt, out, x, 96, 96, HW, bsX, bsX, bsX, 0);

  // ---------------- GDFN ----------------
  layernorm_c<<<(unsigned)((Bn * HW) / 256), 256, 0, stream>>>(out, ln2_g, ln2_b, y2);

  gemm1x1<<<dim3(HW / 512, (2 * HID + 15) / 16, Bn), 256, 0, stream>>>(
      gin_w, y2, g1, nullptr, 2 * HID, 96, HW, bsX, (long)2 * HID * HW, 0, 0);
  dwconv3x3<<<(unsigned)((Bn * 2 * HID * HW) / 256), 256, 0, stream>>>(
      g1, gdw_w, g2, 2 * HID, (long)Bn * 2 * HID * HW);
  gdfn_gate<<<(unsigned)((Bn * HID * HW) / 256), 256, 0, stream>>>(g2, gg);

  // out conv + residual in place: d_out += conv1x1(gg, gout_w)
  gemm1x1<<<dim3(HW / 512, 96 / 16, Bn), 256, 0, stream>>>(
      gout_w, gg, out, out, 96, HID, HW, (long)HID * HW, bsX, bsX, 0);
}